// mgn_GRU_D_cell_60224031424823
// MI455X (gfx1250) — compile-verified
//
#include <hip/hip_runtime.h>
#include <hip/hip_bf16.h>

typedef __attribute__((ext_vector_type(2))) float v2f;
typedef __attribute__((ext_vector_type(4))) float v4f;
typedef __attribute__((ext_vector_type(8))) float v8f;

#define GRUD_B 16384
#define GRUD_I 128
#define GRUD_D 16
#define HDIM   (GRUD_I * GRUD_D)   // 2048
#define TILES_PER_WAVE 8
#define WAVES_PER_BLOCK 8

// sigmoid(x) = 1/(1+exp(-x)) with 1-ULP v_rcp_f32 instead of IEEE divide
__device__ __forceinline__ float fast_sigmoid(float x) {
  return __builtin_amdgcn_rcpf(1.0f + __expf(-x));
}
// tanh(x) = 2*sigmoid(2x) - 1  (one exp + one rcp + one fma)
__device__ __forceinline__ float fast_tanh(float x) {
  const float s = __builtin_amdgcn_rcpf(1.0f + __expf(-2.0f * x));
  return fmaf(2.0f, s, -1.0f);
}
__device__ __forceinline__ void load8(const float* __restrict__ p, float* dst) {
  v4f a = *(const v4f*)(p);
  v4f b = *(const v4f*)(p + 4);
  dst[0] = a.x; dst[1] = a.y; dst[2] = a.z; dst[3] = a.w;
  dst[4] = b.x; dst[5] = b.y; dst[6] = b.z; dst[7] = b.w;
}

// One wave = one feature i, TILES_PER_WAVE batch tiles of 16 rows.
// WMMA mapping (per tile): D[e][b] = sum_d U[d][e] * hh[b][d]
//   A = U^T   (16x16 fp32, A-layout, 4 K-slices of v2f, resident per wave)
//   B = hh^T  (B-layout, derived from C-layout by a lane^16 exchange)
//   C/D: lane = batch col (lane&15), vgpr r + 8*hi = output dim e
__global__ __launch_bounds__(WAVES_PER_BLOCK * 32) void grud_cell_kernel(
    const float* __restrict__ h,  const float* __restrict__ X,
    const float* __restrict__ M,  const float* __restrict__ gamma,
    const float* __restrict__ Wr, const float* __restrict__ Wz, const float* __restrict__ Wh,
    const float* __restrict__ Ur, const float* __restrict__ Uz, const float* __restrict__ Uh,
    const float* __restrict__ Vr, const float* __restrict__ Vz, const float* __restrict__ Vh,
    const float* __restrict__ br, const float* __restrict__ bz, const float* __restrict__ bh,
    float* __restrict__ out)
{
  const int lane = threadIdx.x & 31;
  const int wid  = threadIdx.x >> 5;
  const int col  = lane & 15;      // batch column within tile / N-dim col
  const int hi   = lane >> 4;      // upper-half-lane flag

  const int chunks = GRUD_B / (16 * TILES_PER_WAVE);          // 128
  const int ig     = blockIdx.x / chunks;                     // feature group
  const int chunk  = blockIdx.x % chunks;                     // batch chunk
  const int i      = ig * WAVES_PER_BLOCK + wid;              // this wave's feature
  const int bbase  = chunk * (16 * TILES_PER_WAVE);

  // ---- per-feature constants, resident across the batch-tile loop ----
  const float* Uri = Ur + (long)i * 256;
  const float* Uzi = Uz + (long)i * 256;
  const float* Uhi = Uh + (long)i * 256;

  // A-layout for U^T: slice k, lane L(0-15): A[L][4k..4k+1] = U[4k..4k+1][L]
  //                   lane L(16-31):         A[L-16][4k+2..3] = U[4k+2..3][L-16]
  v2f uaR[4], uaZ[4], uaH[4];
#pragma unroll
  for (int k = 0; k < 4; ++k) {
    const int row = 4 * k + 2 * hi;
    uaR[k].x = Uri[(row    ) * 16 + col];
    uaR[k].y = Uri[(row + 1) * 16 + col];
    uaZ[k].x = Uzi[(row    ) * 16 + col];
    uaZ[k].y = Uzi[(row + 1) * 16 + col];
    uaH[k].x = Uhi[(row    ) * 16 + col];
    uaH[k].y = Uhi[(row + 1) * 16 + col];
  }

  // Per-feature vectors in C-layout order: element r <-> dim d = r + 8*hi
  float wr[8], wz[8], wh[8], vr[8], vz[8], vh[8], cbr[8], cbz[8], cbh[8];
  {
    const long off = (long)i * 16 + 8 * hi;
    load8(Wr + off, wr);  load8(Wz + off, wz);  load8(Wh + off, wh);
    load8(Vr + off, vr);  load8(Vz + off, vz);  load8(Vh + off, vh);
    load8(br + off, cbr); load8(bz + off, cbz); load8(bh + off, cbh);
  }

#pragma unroll 1
  for (int t = 0; t < TILES_PER_WAVE; ++t) {
    const int b0 = bbase + 16 * t;
    const long rowoff = (long)(b0 + col) * HDIM + (long)i * 16 + 8 * hi;
    const float* hp = h + rowoff;       // contiguous d = 8*hi .. 8*hi+7
    const float* gp = gamma + rowoff;

    // hh directly in C-layout: hhC[r] = hh[b=col][d = r + 8*hi]
    float hhC[8];
    {
      v4f h0 = *(const v4f*)(hp);
      v4f h1 = *(const v4f*)(hp + 4);
      v4f g0 = *(const v4f*)(gp);
      v4f g1 = *(const v4f*)(gp + 4);
      hhC[0] = h0.x * g0.x; hhC[1] = h0.y * g0.y;
      hhC[2] = h0.z * g0.z; hhC[3] = h0.w * g0.w;
      hhC[4] = h1.x * g1.x; hhC[5] = h1.y * g1.y;
      hhC[6] = h1.z * g1.z; hhC[7] = h1.w * g1.w;
    }
    const float Xs = X[(long)(b0 + col) * GRUD_I + i];
    const float Ms = M[(long)(b0 + col) * GRUD_I + i];

    // C-layout -> B-layout of hh (lane <-> lane^16 exchange)
    const float u0 = __shfl_xor(hhC[0], 16), u1 = __shfl_xor(hhC[1], 16);
    const float u2 = __shfl_xor(hhC[2], 16), u3 = __shfl_xor(hhC[3], 16);
    const float u4 = __shfl_xor(hhC[4], 16), u5 = __shfl_xor(hhC[5], 16);
    const float u6 = __shfl_xor(hhC[6], 16), u7 = __shfl_xor(hhC[7], 16);
    float hhB[8];
    hhB[0] = hi ? u2     : hhC[0];
    hhB[1] = hi ? u3     : hhC[1];
    hhB[2] = hi ? u6     : hhC[4];
    hhB[3] = hi ? u7     : hhC[5];
    hhB[4] = hi ? hhC[2] : u0;
    hhB[5] = hi ? hhC[3] : u1;
    hhB[6] = hi ? hhC[6] : u4;
    hhB[7] = hi ? hhC[7] : u5;

    // Seed accumulators with X*W + M*V + b directly in C-layout
    v8f accR, accZ;
#pragma unroll
    for (int r = 0; r < 8; ++r) {
      accR[r] = fmaf(Xs, wr[r], fmaf(Ms, vr[r], cbr[r]));
      accZ[r] = fmaf(Xs, wz[r], fmaf(Ms, vz[r], cbz[r]));
    }
    // K=16 via 4 chained 16x16x4 fp32 WMMAs, r and z interleaved
#pragma unroll
    for (int k = 0; k < 4; ++k) {
      v2f b; b.x = hhB[2 * k]; b.y = hhB[2 * k + 1];
      accR = __builtin_amdgcn_wmma_f32_16x16x4_f32(false, uaR[k], false, b,
                                                   (short)0, accR, false, false);
      accZ = __builtin_amdgcn_wmma_f32_16x16x4_f32(false, uaZ[k], false, b,
                                                   (short)0, accZ, false, false);
    }

    float rg[8], zg[8], rhC[8];
#pragma unroll
    for (int r = 0; r < 8; ++r) {
      rg[r]  = fast_sigmoid(accR[r]);
      zg[r]  = fast_sigmoid(accZ[r]);
      rhC[r] = rg[r] * hhC[r];
    }

    // C-layout -> B-layout of (r * hh) to feed the U_h matmul
    const float w0 = __shfl_xor(rhC[0], 16), w1 = __shfl_xor(rhC[1], 16);
    const float w2 = __shfl_xor(rhC[2], 16), w3 = __shfl_xor(rhC[3], 16);
    const float w4 = __shfl_xor(rhC[4], 16), w5 = __shfl_xor(rhC[5], 16);
    const float w6 = __shfl_xor(rhC[6], 16), w7 = __shfl_xor(rhC[7], 16);
    float rhB[8];
    rhB[0] = hi ? w2     : rhC[0];
    rhB[1] = hi ? w3     : rhC[1];
    rhB[2] = hi ? w6     : rhC[4];
    rhB[3] = hi ? w7     : rhC[5];
    rhB[4] = hi ? rhC[2] : w0;
    rhB[5] = hi ? rhC[3] : w1;
    rhB[6] = hi ? rhC[6] : w4;
    rhB[7] = hi ? rhC[7] : w5;

    v8f accH;
#pragma unroll
    for (int r = 0; r < 8; ++r)
      accH[r] = fmaf(Xs, wh[r], fmaf(Ms, vh[r], cbh[r]));
#pragma unroll
    for (int k = 0; k < 4; ++k) {
      v2f b; b.x = rhB[2 * k]; b.y = rhB[2 * k + 1];
      accH = __builtin_amdgcn_wmma_f32_16x16x4_f32(false, uaH[k], false, b,
                                                   (short)0, accH, false, false);
    }

    // out = z*hh + (1-z)*tanh(.)  ==  ht + z*(hh - ht)
    float o[8];
#pragma unroll
    for (int r = 0; r < 8; ++r) {
      const float ht = fast_tanh(accH[r]);
      o[r] = fmaf(zg[r], hhC[r] - ht, ht);
    }
    float* op = out + (long)(b0 + col) * HDIM + (long)i * 16 + 8 * hi;
    v4f s0; s0.x = o[0]; s0.y = o[1]; s0.z = o[2]; s0.w = o[3];
    v4f s1; s1.x = o[4]; s1.y = o[5]; s1.z = o[6]; s1.w = o[7];
    __builtin_nontemporal_store(s0, (v4f*)(op));
    __builtin_nontemporal_store(s1, (v4f*)(op + 4));
  }
}

extern "C" void kernel_launch(void* const* d_in, const int* in_sizes, int n_in,
                              void* d_out, int out_size, void* d_ws, size_t ws_size,
                              hipStream_t stream) {
  const float* h  = (const float*)d_in[0];
  const float* X  = (const float*)d_in[1];
  const float* M  = (const float*)d_in[2];
  const float* g  = (const float*)d_in[3];
  const float* Wr = (const float*)d_in[4];
  const float* Wz = (const float*)d_in[5];
  const float* Wh = (const float*)d_in[6];
  const float* Ur = (const float*)d_in[7];
  const float* Uz = (const float*)d_in[8];
  const float* Uh = (const float*)d_in[9];
  const float* Vr = (const float*)d_in[10];
  const float* Vz = (const float*)d_in[11];
  const float* Vh = (const float*)d_in[12];
  const float* br = (const float*)d_in[13];
  const float* bz = (const float*)d_in[14];
  const float* bh = (const float*)d_in[15];
  float* out = (float*)d_out;

  dim3 grid((GRUD_I / WAVES_PER_BLOCK) * (GRUD_B / (16 * TILES_PER_WAVE)));  // 2048
  dim3 block(WAVES_PER_BLOCK * 32);                                          // 256
  hipLaunchKernelGGL(grud_cell_kernel, grid, block, 0, stream,
                     h, X, M, g, Wr, Wz, Wh, Ur, Uz, Uh, Vr, Vz, Vh, br, bz, bh, out);
}